// ParallelComplexModalFilter_481036337266
// MI455X (gfx1250) — compile-verified
//
#include <hip/hip_runtime.h>
#include <stdint.h>
#include <cmath>

// ParallelComplexModalFilter for MI455X (gfx1250).
//
// h[t,d] = sum_m Re( R(m,d) * p(m,d)^(t-1) ),  p = r*exp(i*theta), t >= 1;
// row t=0 is h_0. Output is h.T: [D=1024, L=4096] fp32 (16 MB).
//
// Roofline: output 16 MB / 23.3 TB/s ~= 0.7 us floor; inputs ~260 KB (L2).
// Algorithm: complex geometric recurrence z <- z*p (4 FMA per mode per step)
// instead of per-(t,m,d) pow/sincos. t is chunked into 64-step tiles for
// parallelism; each chunk seeds z = p^(t0-1) by *uniform* complex
// square-and-multiply (t0-1 is block-uniform, <= 12 bits) -- no large-argument
// sincos, no exp/log, no divergent slow paths.
// Output drains from an LDS tile via global_store_async_from_lds_b128
// (ASYNCcnt, CDNA5-specific) in 128-bit segments, finished by s_wait_asynccnt.
// WMMA does not apply: the m-contraction has both factors varying in (t,m,d).

#define M_MODES 16
#define D_MODEL 1024
#define SEQ_L   4096
#define DTILE   128              // d columns per block == blockDim.x (4 waves)
#define TCHUNK  64               // t steps per block
#define ROW_DW  (TCHUNK + 4)     // padded row stride (dwords): 16B-aligned, bank-skewed

__global__ __launch_bounds__(DTILE)
void modal_filter_kernel(const float* __restrict__ r,
                         const float* __restrict__ theta,
                         const float* __restrict__ Rre,
                         const float* __restrict__ Rim,
                         const float* __restrict__ h0,
                         float* __restrict__ out)
{
    __shared__ float tile[DTILE * ROW_DW];   // 128 * 68 * 4 = 34,816 B

    const int tid = threadIdx.x;
    const int t0  = blockIdx.x * TCHUNK;
    const int d0  = blockIdx.y * DTILE;
    const int d   = d0 + tid;

    // Per-mode complex state: z = p^(t-1) for the current t; coefficients R.
    float zre[M_MODES], zim[M_MODES];
    float pre_[M_MODES], pim_[M_MODES];
    float cre[M_MODES], cim[M_MODES];

    // Column t uses exponent e = t-1 (t=0 is the h_0 row). Seed at e0 (uniform).
    const unsigned e0 = (t0 == 0) ? 0u : (unsigned)(t0 - 1);

    #pragma unroll
    for (int m = 0; m < M_MODES; ++m) {
        const int idx  = m * D_MODEL + d;
        const float rv = r[idx];
        const float th = theta[idx];
        cre[m] = Rre[idx];
        cim[m] = Rim[idx];

        float s, c;
        __sincosf(th, &s, &c);                // th in [0, 2pi): fast path only
        const float pr = rv * c;
        const float pi = rv * s;
        pre_[m] = pr;
        pim_[m] = pi;

        // z = p^e0 via square-and-multiply; e0 is block-uniform (no divergence).
        // ~24 complex mults worst case; r^e0 underflow matches fp32 reference.
        float zr = 1.0f, zi = 0.0f;
        float br = pr, bi = pi;
        unsigned e = e0;
        while (e) {
            if (e & 1u) {
                const float nr = zr * br - zi * bi;
                zi = zr * bi + zi * br;
                zr = nr;
            }
            const float sr = br * br - bi * bi;
            bi = 2.0f * br * bi;
            br = sr;
            e >>= 1;
        }
        zre[m] = zr;
        zim[m] = zi;
    }

    float* row = &tile[tid * ROW_DW];
    #pragma unroll 2
    for (int k = 0; k < TCHUNK; ++k) {
        const int t = t0 + k;
        float val;
        if (t == 0) {
            val = h0[d];                      // prepended h_0 row (z stays p^0 for t=1)
        } else {
            val = 0.0f;
            #pragma unroll
            for (int m = 0; m < M_MODES; ++m)
                val += cre[m] * zre[m] - cim[m] * zim[m];
            #pragma unroll
            for (int m = 0; m < M_MODES; ++m) {
                const float nr = zre[m] * pre_[m] - zim[m] * pim_[m];
                const float ni = zre[m] * pim_[m] + zim[m] * pre_[m];
                zre[m] = nr;
                zim[m] = ni;
            }
        }
        row[k] = val;                         // ds_store_b32, <=2-way bank conflict
    }
    __syncthreads();

    // Drain tile -> out[d][t] via async LDS->global DMA, 16B per lane per issue.
    // 16 consecutive lanes cover one contiguous 256B global row span.
    const uint32_t lds0 = (uint32_t)(uintptr_t)(&tile[0]);  // low 32 bits == LDS offset
    #pragma unroll
    for (int i = 0; i < TCHUNK / 4; ++i) {    // 16 issues/thread, 2048 segments total
        const int seg  = i * DTILE + tid;
        const int rowi = seg >> 4;            // TCHUNK/4 == 16 segments per row
        const int coli = seg & 15;
        const uint32_t laddr = lds0 + (uint32_t)(rowi * ROW_DW + coli * 4) * 4u;
        const float* gp = out + (size_t)(d0 + rowi) * SEQ_L + (size_t)(t0 + coli * 4);
        const uint64_t ga = (uint64_t)(uintptr_t)gp;
        asm volatile("global_store_async_from_lds_b128 %0, %1, off"
                     :: "v"(ga), "v"(laddr)
                     : "memory");
    }
    asm volatile("s_wait_asynccnt 0" ::: "memory");
}

extern "C" void kernel_launch(void* const* d_in, const int* in_sizes, int n_in,
                              void* d_out, int out_size, void* d_ws, size_t ws_size,
                              hipStream_t stream) {
    // setup_inputs order: L (int scalar; 4096 at compile time), r, theta, R_re, R_im, h_0
    (void)in_sizes; (void)n_in; (void)out_size; (void)d_ws; (void)ws_size;
    const float* r     = (const float*)d_in[1];
    const float* theta = (const float*)d_in[2];
    const float* Rre   = (const float*)d_in[3];
    const float* Rim   = (const float*)d_in[4];
    const float* h0    = (const float*)d_in[5];
    float* out = (float*)d_out;

    dim3 grid(SEQ_L / TCHUNK, D_MODEL / DTILE);   // 64 x 8 = 512 blocks, 128 thr each
    modal_filter_kernel<<<grid, DTILE, 0, stream>>>(r, theta, Rre, Rim, h0, out);
}